// MultiHeadAttentionWithCache_18262200942834
// MI455X (gfx1250) — compile-verified
//
#include <hip/hip_runtime.h>
#include <cstdint>

#define B_   16
#define SQ_  16
#define SKC_ 4080
#define SK_  4096
#define NS_  1024
#define NH_  16
#define D_   64
#define NEG_INF (-__builtin_inff())

typedef __attribute__((ext_vector_type(2))) float v2f;
typedef __attribute__((ext_vector_type(8))) float v8f;

// The async-to-LDS builtin takes int4-vector pointers:
//   param0: v4i __device__* (AS1), param1: v4i __shared__* (AS3), imm, imm
typedef __attribute__((__vector_size__(16))) int v4i_t;
typedef __attribute__((address_space(1))) v4i_t GV4;
typedef __attribute__((address_space(3))) v4i_t LV4;

#if __has_builtin(__builtin_amdgcn_global_load_async_to_lds_b128)
#define HAVE_ASYNC_LDS 1
#define ASYNC_COPY16(dst, src)                                                 \
  __builtin_amdgcn_global_load_async_to_lds_b128((GV4*)(src), (LV4*)(dst), 0, 0)
#endif

static __device__ __forceinline__ void async_wait0() {
#if __has_builtin(__builtin_amdgcn_s_wait_asynccnt)
  __builtin_amdgcn_s_wait_asynccnt(0);
#else
  asm volatile("s_wait_asynccnt 0" ::: "memory");
#endif
}

static __device__ __forceinline__ v8f wmma_f32(v2f a, v2f b, v8f c) {
  // D = A(16x4) * B(4x16) + C, fp32 everywhere -> v_wmma_f32_16x16x4_f32
  return __builtin_amdgcn_wmma_f32_16x16x4_f32(false, a, false, b, (short)0, c,
                                               false, false);
}

// ---------------------------------------------------------------------------
// 1) Stream cached K/V rows into the output buffers (the dominant HBM cost).
// ---------------------------------------------------------------------------
__global__ void __launch_bounds__(256)
copy_cache(const float4* __restrict__ kc, const float4* __restrict__ vc,
           float4* __restrict__ kd, float4* __restrict__ vd) {
  const size_t i = (size_t)blockIdx.x * 256 + threadIdx.x;   // < 16*4080*256
  const size_t per_b = (size_t)SKC_ * (NS_ / 4);
  const size_t b = i / per_b;
  const size_t rem = i - b * per_b;
  const size_t dst = b * (size_t)SK_ * (NS_ / 4) + rem;
  kd[dst] = kc[i];
  vd[dst] = vc[i];
}

// ---------------------------------------------------------------------------
// 2) Generic 256x1024x1024 fp32 WMMA GEMM:  out = (A @ W + bias) * alpha
//    Output row mapping supports writing into the tail of the KV buffers:
//      gm = global row; out[(gm>>4)*out_bstride + (tail_off + (gm&15))*NS + n]
// ---------------------------------------------------------------------------
__global__ void __launch_bounds__(256)
gemm256_wmma(const float* __restrict__ A, const float* __restrict__ W,
             const float* __restrict__ bias, float* __restrict__ out,
             long out_bstride, int tail_off, float alpha) {
  __shared__ float sA[16][1028];                 // 16-row A tile, padded
  const int tid  = threadIdx.x;
  const int lane = tid & 31;
  const int w    = tid >> 5;
  const int m0   = blockIdx.x << 4;
  const int n0   = (blockIdx.y << 7) + (w << 4);

  // Stage A tile: async direct-to-LDS when available (no VGPR round trip).
#ifdef HAVE_ASYNC_LDS
  for (int i = tid; i < 16 * 256; i += 256) {
    const int r = i >> 8, c4 = i & 255;
    ASYNC_COPY16(&sA[r][c4 << 2], A + (size_t)(m0 + r) * NS_ + (c4 << 2));
  }
  async_wait0();
#else
  for (int i = tid; i < 16 * 256; i += 256) {
    const int r = i >> 8, c4 = i & 255;
    const float4 val =
        reinterpret_cast<const float4*>(A)[(size_t)(m0 + r) * (NS_ / 4) + c4];
    *reinterpret_cast<float4*>(&sA[r][c4 << 2]) = val;
  }
#endif
  __syncthreads();

  const int mrow = lane & 15;                    // A: M row / B: N col
  const int koff = (lane >> 4) << 1;             // lane-half K pair
  v8f acc = {};
#pragma unroll 4
  for (int kk = 0; kk < NS_; kk += 4) {
    v2f a;
    a.x = sA[mrow][kk + koff];
    a.y = sA[mrow][kk + koff + 1];
    const float* wp = W + (size_t)(kk + koff) * NS_ + n0 + mrow;
    v2f b;
    b.x = wp[0];                                 // K = kk+koff
    b.y = wp[NS_];                               // K = kk+koff+1
    acc = wmma_f32(a, b, acc);
  }

  const int ncol  = n0 + mrow;
  const float bvl = bias ? bias[ncol] : 0.0f;
  const int rbase = (lane >> 4) << 3;            // C/D: rows r / r+8 per half
#pragma unroll
  for (int r = 0; r < 8; ++r) {
    const int gm = m0 + r + rbase;
    out[(size_t)(gm >> 4) * out_bstride +
        (size_t)(tail_off + (gm & 15)) * NS_ + ncol] = (acc[r] + bvl) * alpha;
  }
}

// ---------------------------------------------------------------------------
// 3) Attention: one workgroup per (b,h). Online (flash) softmax over Sk=4096
//    in chunks of 256 columns. Scores via WMMA by all 8 waves; P@V
//    accumulated by waves 0..3 (one 16x16 N-tile of the 16x64 output each).
//    Next chunk's K/V rows are prefetched while the current chunk computes.
// ---------------------------------------------------------------------------
__global__ void __launch_bounds__(256)
attn_wmma(const float* __restrict__ q, const float* __restrict__ kmat,
          const float* __restrict__ vmat, float* __restrict__ wv) {
  __shared__ float sQ[16][68];                   // q tile (pre-scaled), padded
  __shared__ float sS[16][260];                  // score / P chunk, padded
  __shared__ float sRed[16][16];
  __shared__ float sM[16], sL[16], sCf[16];

  const int tid  = threadIdx.x;
  const int lane = tid & 31;
  const int w    = tid >> 5;
  const int b    = blockIdx.x >> 4;
  const int h    = blockIdx.x & 15;

  const float* qb = q + (size_t)(b * SQ_) * NS_ + h * D_;
#ifdef HAVE_ASYNC_LDS
  {
    const int r = tid >> 4, c4 = tid & 15;       // 256 x 16B = whole tile
    ASYNC_COPY16(&sQ[r][c4 << 2], qb + (size_t)r * NS_ + (c4 << 2));
    async_wait0();
  }
#else
  for (int i = tid; i < SQ_ * D_; i += 256) {
    const int r = i >> 6, cc = i & 63;
    sQ[r][cc] = qb[(size_t)r * NS_ + cc];
  }
#endif
  if (tid < 16) { sM[tid] = NEG_INF; sL[tid] = 0.0f; }
  __syncthreads();

  const float* Kb = kmat + (size_t)b * SK_ * NS_ + h * D_;
  const float* Vb = vmat + (size_t)b * SK_ * NS_ + h * D_;
  const int mrow  = lane & 15;
  const int koff  = (lane >> 4) << 1;
  const int rbase = (lane >> 4) << 3;
  const int row   = tid >> 4, sub = tid & 15;
  v8f acc = {};                                  // live in waves 0..3

  for (int c = 0; c < SK_ / 256; ++c) {
    const int col0 = c << 8;
    const bool more = (c + 1) < (SK_ / 256);

    // ---- scores: wave w computes tiles t = w, w+8 ----
    for (int t = w; t < 16; t += 8) {
      const int n0 = col0 + (t << 4);
      v8f s = {};
      const float* kp = Kb + (size_t)(n0 + mrow) * NS_ + koff; // lane = N col
      if (more) {                                // prefetch same rows, chunk c+1
        __builtin_prefetch(kp + 256 * NS_, 0, 1);
        __builtin_prefetch(kp + 256 * NS_ + 32, 0, 1);
      }
#pragma unroll 4
      for (int kk = 0; kk < D_; kk += 4) {
        v2f a;
        a.x = sQ[mrow][kk + koff];
        a.y = sQ[mrow][kk + koff + 1];
        v2f bb;
        bb.x = kp[kk];
        bb.y = kp[kk + 1];
        s = wmma_f32(a, bb, s);
      }
      const int ncol = n0 + mrow;
#pragma unroll
      for (int r = 0; r < 8; ++r) {
        float val = s[r];
        const int mr = r + rbase;
        if (ncol < SQ_ && ncol > mr) val = NEG_INF;   // triu mask, first Sq cols
        sS[mr][(t << 4) + mrow] = val;
      }
    }
    __syncthreads();

    // ---- online softmax stats ----
    float lmax = NEG_INF;
#pragma unroll
    for (int j = 0; j < 16; ++j) lmax = fmaxf(lmax, sS[row][(sub << 4) + j]);
    sRed[row][sub] = lmax;
    __syncthreads();
    if (tid < 16) {
      float cmax = NEG_INF;
      for (int j = 0; j < 16; ++j) cmax = fmaxf(cmax, sRed[tid][j]);
      const float mOld = sM[tid];
      const float mNew = fmaxf(mOld, cmax);
      sCf[tid] = (mOld == NEG_INF) ? 0.0f : __expf(mOld - mNew);
      sM[tid]  = mNew;
    }
    __syncthreads();
    const float mNew = sM[row];
    float lsum = 0.0f;
#pragma unroll
    for (int j = 0; j < 16; ++j) {
      const float p = __expf(sS[row][(sub << 4) + j] - mNew);
      sS[row][(sub << 4) + j] = p;
      lsum += p;
    }
    sRed[row][sub] = lsum;
    __syncthreads();
    if (tid < 16) {
      float sum = 0.0f;
      for (int j = 0; j < 16; ++j) sum += sRed[tid][j];
      sL[tid] = sL[tid] * sCf[tid] + sum;
    }
    __syncthreads();

    // ---- P @ V accumulation: waves 0..3, one 16-col D tile each ----
    if (w < 4) {
#pragma unroll
      for (int r = 0; r < 8; ++r) acc[r] *= sCf[r + rbase];
      const float* vp0 = Vb + (size_t)(col0 + koff) * NS_ + (w << 4) + mrow;
#pragma unroll 4
      for (int kk = 0; kk < 256; kk += 4) {
        if ((kk & 31) == 0 && more)              // prefetch V rows, chunk c+1
          __builtin_prefetch(
              Vb + (size_t)(col0 + 256 + kk + lane) * NS_ + (w << 4), 0, 1);
        v2f a;
        a.x = sS[mrow][kk + koff];
        a.y = sS[mrow][kk + koff + 1];
        const float* vp = vp0 + (size_t)kk * NS_;
        v2f bb;
        bb.x = vp[0];                            // K = col0+kk+koff
        bb.y = vp[NS_];                          // K = col0+kk+koff+1
        acc = wmma_f32(a, bb, acc);
      }
    }
    __syncthreads();                             // sS reused next chunk
  }

  if (w < 4) {
#pragma unroll
    for (int r = 0; r < 8; ++r) {
      const int mr = r + rbase;
      wv[(size_t)(b * SQ_ + mr) * NS_ + h * D_ + (w << 4) + mrow] =
          acc[r] / sL[mr];
    }
  }
}

// ---------------------------------------------------------------------------
// Launcher
// ---------------------------------------------------------------------------
extern "C" void kernel_launch(void* const* d_in, const int* in_sizes, int n_in,
                              void* d_out, int out_size, void* d_ws,
                              size_t ws_size, hipStream_t stream) {
  const float* x  = (const float*)d_in[0];
  const float* kc = (const float*)d_in[1];
  const float* vc = (const float*)d_in[2];
  const float* Wq = (const float*)d_in[3];
  const float* bq = (const float*)d_in[4];
  const float* Wk = (const float*)d_in[5];
  const float* Wv = (const float*)d_in[6];
  const float* bv = (const float*)d_in[7];
  const float* Wo = (const float*)d_in[8];
  const float* bo = (const float*)d_in[9];

  float* out  = (float*)d_out;
  float* kout = out + (size_t)B_ * SQ_ * NS_;          // 262144
  float* vout = kout + (size_t)B_ * SK_ * NS_;         // + 67108864
  float* q_ws  = (float*)d_ws;                         // 1 MB
  float* wv_ws = q_ws + (size_t)B_ * SQ_ * NS_;        // 1 MB

  // 1) cache copy: 16*4080*256 float4 per tensor
  copy_cache<<<65280, 256, 0, stream>>>((const float4*)kc, (const float4*)vc,
                                        (float4*)kout, (float4*)vout);

  // 2) projections (q pre-scaled by D^-1/2 so attention q-load is a pure copy)
  dim3 g(16, 8);
  gemm256_wmma<<<g, 256, 0, stream>>>(x, Wq, bq, q_ws, (long)SQ_ * NS_, 0, 0.125f);
  gemm256_wmma<<<g, 256, 0, stream>>>(x, Wk, nullptr, kout, (long)SK_ * NS_, SKC_, 1.0f);
  gemm256_wmma<<<g, 256, 0, stream>>>(x, Wv, bv, vout, (long)SK_ * NS_, SKC_, 1.0f);

  // 3) attention
  attn_wmma<<<B_ * NH_, 256, 0, stream>>>(q_ws, kout, vout, wv_ws);

  // 4) output projection
  gemm256_wmma<<<g, 256, 0, stream>>>(wv_ws, Wo, bo, out, (long)SQ_ * NS_, 0, 1.0f);
}